// GCN_31662498906818
// MI455X (gfx1250) — compile-verified
//
#include <hip/hip_runtime.h>
#include <math.h>

// Problem constants (match reference)
constexpr int GG = 32;      // graphs
constexpr int NN = 512;     // nodes per graph
constexpr int EE = 8192;    // edges per graph
constexpr int DD = 256;     // feature dim
constexpr int KK = 2048;    // codebook entries
constexpr int MTOT = GG * NN;           // 16384 rows
constexpr float ALPHA = 100.0f;
constexpr float LAMB_EDGE = 1.0f;
constexpr float CODEBOOK_WEIGHT = 1000.0f;
constexpr int TPG = (NN / 16) * (NN / 16);   // 1024 logit tiles per graph

typedef __attribute__((ext_vector_type(2))) float v2f;
typedef __attribute__((ext_vector_type(8))) float v8f;

__device__ __forceinline__ float softplusf(float x) {
    return fmaxf(x, 0.0f) + log1pf(expf(-fabsf(x)));   // numerically stable
}

// ---------------------------------------------------------------- utilities
__global__ void zero_f32_kernel(float* p, long n) {
    long i = (long)blockIdx.x * blockDim.x + threadIdx.x;
    long s = (long)gridDim.x * blockDim.x;
    for (; i < n; i += s) p[i] = 0.0f;
}

__global__ void zero_u8_kernel(unsigned char* p, long n) {
    long i = (long)blockIdx.x * blockDim.x + threadIdx.x;
    long s = (long)gridDim.x * blockDim.x;
    for (; i < n; i += s) p[i] = 0;
}

// ------------------------------------------------------------- degree + rs
__global__ void deg_kernel(const int* src, const int* dst, float* dout, int* din) {
    int t = blockIdx.x * blockDim.x + threadIdx.x;
    if (t >= GG * EE) return;
    int g = t / EE, e = t % EE;
    atomicAdd(&dout[g * NN + src[g * EE + e]], 1.0f);   // integer-valued: exact
    atomicAdd(&din [g * NN + dst[g * EE + e]], 1);
}

__global__ void rs_kernel(float* d, int n) {   // in-place rsqrt(clip(deg,1))
    int i = blockIdx.x * blockDim.x + threadIdx.x;
    if (i < n) d[i] = rsqrtf(fmaxf(d[i], 1.0f));
}

__global__ void rs_int_kernel(const int* d, float* out, int n) {
    int i = blockIdx.x * blockDim.x + threadIdx.x;
    if (i < n) out[i] = rsqrtf(fmaxf((float)d[i], 1.0f));
}

// ----------------------------------------------------------- CSR construction
// Exclusive prefix sum of in-degrees per graph (serial scan: 512 elems, 32 blocks)
__global__ void scan_kernel(const int* degin, int* rowstart, int* cursor) {
    __shared__ int s[NN];
    int g = blockIdx.x;
    if (threadIdx.x == 0) {
        int run = g * EE;
        for (int n = 0; n < NN; ++n) { s[n] = run; run += degin[g * NN + n]; }
    }
    __syncthreads();
    for (int n = threadIdx.x; n < NN; n += blockDim.x) {
        rowstart[g * NN + n] = s[n];
        cursor  [g * NN + n] = s[n];
    }
}

__global__ void csr_fill_kernel(const int* src, const int* dst, int* cursor, int* csrsrc) {
    int t = blockIdx.x * blockDim.x + threadIdx.x;
    if (t >= GG * EE) return;
    int g = t / EE, e = t % EE;
    int d = dst[g * EE + e];
    int pos = atomicAdd(&cursor[g * NN + d], 1);
    csrsrc[pos] = src[g * EE + e];
}

// Sort each node's source list so the gather sum order is deterministic.
__global__ void sort_csr_kernel(const int* rowstart, const int* degin, int* csrsrc) {
    int t = blockIdx.x * blockDim.x + threadIdx.x;
    if (t >= MTOT) return;
    int beg = rowstart[t], cnt = degin[t];
    for (int i = 1; i < cnt; ++i) {
        int key = csrsrc[beg + i];
        int j = i - 1;
        while (j >= 0 && csrsrc[beg + j] > key) { csrsrc[beg + j + 1] = csrsrc[beg + j]; --j; }
        csrsrc[beg + j + 1] = key;
    }
}

// Gather aggregation: m[g,n,:] = sum_e x[g,src_e,:] * rs_out[g,src_e]
__global__ void gather_agg_kernel(const float* __restrict__ x, const int* __restrict__ csrsrc,
                                  const int* __restrict__ rowstart, const int* __restrict__ degin,
                                  const float* __restrict__ rs_out, float* __restrict__ m) {
    int g = blockIdx.x / NN, n = blockIdx.x % NN;
    int t = threadIdx.x;   // feature index
    int beg = rowstart[g * NN + n];
    int cnt = degin[g * NN + n];
    float acc = 0.0f;
    for (int e = 0; e < cnt; ++e) {
        int s = csrsrc[beg + e];
        acc += x[((long)g * NN + s) * DD + t] * rs_out[g * NN + s];
    }
    m[((long)g * NN + n) * DD + t] = acc;
}

// ------------------------------------------------------------ WMMA GEMM
// out[M,256] = act( (gather/scale A)[M,256] @ W[256,256] + bias )
// One wave computes a 32x64 tile: 2 row-frags x 4 col-frags (8 independent WMMA chains).
__global__ void gemm32x64_kernel(const float* __restrict__ A, const int* __restrict__ rowidx,
                                 const float* __restrict__ rowscale,
                                 const float* __restrict__ W, const float* __restrict__ bias,
                                 float* __restrict__ out, int M, int relu) {
    int wave = blockIdx.x * (blockDim.x >> 5) + (threadIdx.x >> 5);
    int lane = threadIdx.x & 31;
    const int CG = DD / 64;                  // 4 column groups
    int total = (M / 32) * CG;
    if (wave >= total) return;               // wave-uniform
    int tm = wave / CG, tg = wave % CG;
    int r = lane & 15, half = lane >> 4;
    int row0 = tm * 32 + r, row1 = row0 + 16;
    long ar0 = rowidx ? (long)rowidx[row0] * DD : (long)row0 * DD;
    long ar1 = rowidx ? (long)rowidx[row1] * DD : (long)row1 * DD;
    float sc0 = rowscale ? rowscale[row0] : 1.0f;
    float sc1 = rowscale ? rowscale[row1] : 1.0f;
    int cn0 = tg * 64;

    v8f acc[2][4];
    #pragma unroll
    for (int rt = 0; rt < 2; ++rt)
        #pragma unroll
        for (int c = 0; c < 4; ++c) acc[rt][c] = (v8f){0.f,0.f,0.f,0.f,0.f,0.f,0.f,0.f};

    for (int k0 = 0; k0 < DD; k0 += 4) {
        int k = k0 + 2 * half;
        v2f a0, a1, b[4];
        a0.x = A[ar0 + k] * sc0;  a0.y = A[ar0 + k + 1] * sc0;
        a1.x = A[ar1 + k] * sc1;  a1.y = A[ar1 + k + 1] * sc1;
        #pragma unroll
        for (int c = 0; c < 4; ++c) {
            b[c].x = W[(long)k       * DD + cn0 + c * 16 + r];
            b[c].y = W[(long)(k + 1) * DD + cn0 + c * 16 + r];
        }
        #pragma unroll
        for (int c = 0; c < 4; ++c) {
            acc[0][c] = __builtin_amdgcn_wmma_f32_16x16x4_f32(false, a0, false, b[c],
                                                              (short)0, acc[0][c], false, false);
            acc[1][c] = __builtin_amdgcn_wmma_f32_16x16x4_f32(false, a1, false, b[c],
                                                              (short)0, acc[1][c], false, false);
        }
    }
    #pragma unroll
    for (int c = 0; c < 4; ++c) {
        float bb = bias[cn0 + c * 16 + r];
        #pragma unroll
        for (int rt = 0; rt < 2; ++rt)
            #pragma unroll
            for (int v = 0; v < 8; ++v) {
                float val = acc[rt][c][v] + bb;
                if (relu) val = fmaxf(val, 0.0f);
                out[(long)(tm * 32 + rt * 16 + v + 8 * half) * DD + cn0 + c * 16 + r] = val;
            }
    }
}

// ------------------------------------------------------------- layernorm
__global__ void layernorm_kernel(float* x, const float* gamma, const float* beta) {
    __shared__ float red[256];
    int row = blockIdx.x, t = threadIdx.x;
    float v = x[(long)row * DD + t];
    red[t] = v; __syncthreads();
    for (int s = 128; s > 0; s >>= 1) { if (t < s) red[t] += red[t + s]; __syncthreads(); }
    float mu = red[0] / DD; __syncthreads();
    float d = v - mu;
    red[t] = d * d; __syncthreads();
    for (int s = 128; s > 0; s >>= 1) { if (t < s) red[t] += red[t + s]; __syncthreads(); }
    float var = red[0] / DD;
    x[(long)row * DD + t] = d * rsqrtf(var + 1e-5f) * gamma[t] + beta[t];
}

// --------------------------------------------------------- L2 row normalize
__global__ void rownorm_kernel(const float* x, float* out) {
    __shared__ float red[256];
    int row = blockIdx.x, t = threadIdx.x;
    float v = x[(long)row * DD + t];
    red[t] = v * v; __syncthreads();
    for (int s = 128; s > 0; s >>= 1) { if (t < s) red[t] += red[t + s]; __syncthreads(); }
    out[(long)row * DD + t] = v * rsqrtf(red[0] + 1e-12f);
}

// ------------------------------------------------- WMMA cosine-sim argmax
// dist = xn[16384,256] @ cn^T[256,2048]; argmax over K per row.
// One wave owns 32 rows; sweeps K in 64-wide groups (8 WMMA chains per pass).
__global__ void vq_argmax_kernel(const float* __restrict__ xn, const float* __restrict__ cn,
                                 int* __restrict__ idx) {
    int wave = blockIdx.x * (blockDim.x >> 5) + (threadIdx.x >> 5);
    int lane = threadIdx.x & 31;
    if (wave >= MTOT / 32) return;
    int r = lane & 15, half = lane >> 4;
    long ar0 = (long)(wave * 32 + r) * DD;
    long ar1 = ar0 + (long)16 * DD;

    float best[2][8]; int bidx[2][8];
    #pragma unroll
    for (int rt = 0; rt < 2; ++rt)
        #pragma unroll
        for (int v = 0; v < 8; ++v) { best[rt][v] = -1e30f; bidx[rt][v] = 0; }

    for (int n0 = 0; n0 < KK; n0 += 64) {
        v8f acc[2][4];
        #pragma unroll
        for (int rt = 0; rt < 2; ++rt)
            #pragma unroll
            for (int c = 0; c < 4; ++c) acc[rt][c] = (v8f){0.f,0.f,0.f,0.f,0.f,0.f,0.f,0.f};
        long brow[4];
        #pragma unroll
        for (int c = 0; c < 4; ++c) brow[c] = (long)(n0 + c * 16 + r) * DD + 2 * half;

        for (int k0 = 0; k0 < DD; k0 += 4) {
            int k = k0 + 2 * half;
            v2f a0, a1, b[4];
            a0.x = xn[ar0 + k];  a0.y = xn[ar0 + k + 1];
            a1.x = xn[ar1 + k];  a1.y = xn[ar1 + k + 1];
            #pragma unroll
            for (int c = 0; c < 4; ++c) { b[c].x = cn[brow[c] + k0]; b[c].y = cn[brow[c] + k0 + 1]; }
            #pragma unroll
            for (int c = 0; c < 4; ++c) {
                acc[0][c] = __builtin_amdgcn_wmma_f32_16x16x4_f32(false, a0, false, b[c],
                                                                  (short)0, acc[0][c], false, false);
                acc[1][c] = __builtin_amdgcn_wmma_f32_16x16x4_f32(false, a1, false, b[c],
                                                                  (short)0, acc[1][c], false, false);
            }
        }
        // candidate index for a lane's value in col-frag c is n0 + c*16 + (lane&15)
        #pragma unroll
        for (int c = 0; c < 4; ++c) {
            int cand = n0 + c * 16 + r;
            #pragma unroll
            for (int rt = 0; rt < 2; ++rt)
                #pragma unroll
                for (int v = 0; v < 8; ++v) {
                    float d = acc[rt][c][v];
                    bool upd = d > best[rt][v];
                    best[rt][v] = upd ? d : best[rt][v];
                    bidx[rt][v] = upd ? cand : bidx[rt][v];
                }
        }
    }
    // reduce across the 16 lanes of each half (rows differ by half); ties -> lowest index
    #pragma unroll
    for (int off = 1; off < 16; off <<= 1) {
        #pragma unroll
        for (int rt = 0; rt < 2; ++rt)
            #pragma unroll
            for (int v = 0; v < 8; ++v) {
                float ov = __shfl_xor(best[rt][v], off, 32);
                int   oi = __shfl_xor(bidx[rt][v], off, 32);
                bool upd = (ov > best[rt][v]) || (ov == best[rt][v] && oi < bidx[rt][v]);
                best[rt][v] = upd ? ov : best[rt][v];
                bidx[rt][v] = upd ? oi : bidx[rt][v];
            }
    }
    if (r == 0) {
        #pragma unroll
        for (int rt = 0; rt < 2; ++rt)
            #pragma unroll
            for (int v = 0; v < 8; ++v)
                idx[wave * 32 + rt * 16 + v + 8 * half] = bidx[rt][v];
    }
}

// ------------------------------------------------------------- VQ loss sum
__global__ void vq_loss_kernel(const float* hf, const float* cb, const int* idx, float* part) {
    __shared__ float red[256];
    long t = (long)blockIdx.x * blockDim.x + threadIdx.x;
    float v = 0.0f;
    if (t < (long)MTOT * DD) {
        int row = (int)(t / DD), d = (int)(t % DD);
        float diff = cb[(long)idx[row] * DD + d] - hf[t];
        v = diff * diff;
    }
    red[threadIdx.x] = v; __syncthreads();
    for (int s = 128; s > 0; s >>= 1) { if (threadIdx.x < s) red[threadIdx.x] += red[threadIdx.x + s]; __syncthreads(); }
    if (threadIdx.x == 0) part[blockIdx.x] = red[0];   // deterministic slot
}

// ------------------------------------------------------------ adjacency
__global__ void adj_build_kernel(const int* src, const int* dst, unsigned char* adj) {
    int t = blockIdx.x * blockDim.x + threadIdx.x;
    if (t >= GG * EE) return;
    int g = t / EE, e = t % EE;
    adj[(long)g * NN * NN + (long)src[g * EE + e] * NN + dst[g * EE + e]] = 1;
}

__global__ void count_edges_kernel(const unsigned char* adj, float* ne) {
    __shared__ float red[256];
    int g = blockIdx.x / NN, i = blockIdx.x % NN;
    int t = threadIdx.x;
    float c = 0.0f;
    for (int j = t; j < NN; j += 256)
        if (j > i) c += (float)adj[(long)g * NN * NN + (long)i * NN + j];
    red[t] = c; __syncthreads();
    for (int s = 128; s > 0; s >>= 1) { if (t < s) red[t] += red[t + s]; __syncthreads(); }
    if (t == 0 && red[0] != 0.0f) atomicAdd(&ne[g], red[0]);   // integer-valued: exact
}

// -------------------------------------------- WMMA edge reconstruction loss
__global__ void edge_loss_kernel(const float* __restrict__ qe, const unsigned char* __restrict__ adj,
                                 const float* __restrict__ ne, float* __restrict__ part) {
    int wave = blockIdx.x * (blockDim.x >> 5) + (threadIdx.x >> 5);
    int lane = threadIdx.x & 31;
    if (wave >= GG * TPG) return;
    int g = wave / TPG, t = wave % TPG;
    int tm = t / (NN / 16), tn = t % (NN / 16);
    if (tn < tm) return;                     // tile fully below diagonal (wave-uniform)
    int r = lane & 15, half = lane >> 4;
    const float* q = qe + (long)g * NN * DD;
    long arow = (long)(tm * 16 + r) * DD;
    long brow = (long)(tn * 16 + r) * DD + 2 * half;

    v8f acc = {0.f,0.f,0.f,0.f,0.f,0.f,0.f,0.f};
    for (int k0 = 0; k0 < DD; k0 += 4) {
        v2f a, b;
        a.x = q[arow + k0 + 2 * half];
        a.y = q[arow + k0 + 2 * half + 1];
        b.x = q[brow + k0];
        b.y = q[brow + k0 + 1];
        acc = __builtin_amdgcn_wmma_f32_16x16x4_f32(false, a, false, b,
                                                    (short)0, acc, false, false);
    }
    float nef = ne[g];
    float pw = ((float)NN * NN * 0.5f - nef) / (nef + 1e-6f);
    float sum = 0.0f;
    int j = tn * 16 + r;
    #pragma unroll
    for (int v = 0; v < 8; ++v) {
        int i = tm * 16 + v + 8 * half;
        if (j > i) {
            float l = acc[v];
            float a8 = (float)adj[(long)g * NN * NN + (long)i * NN + j];
            sum += pw * a8 * softplusf(-l) + (1.0f - a8) * softplusf(l);
        }
    }
    #pragma unroll
    for (int off = 1; off < 32; off <<= 1) sum += __shfl_xor(sum, off, 32);
    if (lane == 0) part[g * TPG + t] = sum;   // deterministic slot (pre-zeroed)
}

// --------------------------------------------------------- reductions
__global__ void reduce_sum_kernel(const float* p, long n, float* out) {   // single block
    __shared__ float red[256];
    float acc = 0.0f;
    for (long i = threadIdx.x; i < n; i += 256) acc += p[i];
    red[threadIdx.x] = acc; __syncthreads();
    for (int s = 128; s > 0; s >>= 1) { if (threadIdx.x < s) red[threadIdx.x] += red[threadIdx.x + s]; __syncthreads(); }
    if (threadIdx.x == 0) out[0] = red[0];
}

__global__ void reduce_graph_kernel(const float* p, float* esum) {        // block per graph
    __shared__ float red[256];
    int g = blockIdx.x;
    float acc = 0.0f;
    for (int i = threadIdx.x; i < TPG; i += 256) acc += p[g * TPG + i];
    red[threadIdx.x] = acc; __syncthreads();
    for (int s = 128; s > 0; s >>= 1) { if (threadIdx.x < s) red[threadIdx.x] += red[threadIdx.x + s]; __syncthreads(); }
    if (threadIdx.x == 0) esum[g] = red[0];
}

// ----------------------------------------------------------------- combine
__global__ void final_kernel(const float* esum, const float* sq, float* out) {
    if (blockIdx.x == 0 && threadIdx.x == 0) {
        float e = 0.0f;
        for (int g = 0; g < GG; ++g) e += esum[g];
        e /= (float)GG * ((float)NN * (NN - 1) * 0.5f);        // mean edge loss
        float vq = CODEBOOK_WEIGHT * sq[0] / ((float)MTOT * DD);
        out[0] = ALPHA * LAMB_EDGE * e + vq;                   // feature term has weight 0
    }
}

// ================================================================= launch
extern "C" void kernel_launch(void* const* d_in, const int* in_sizes, int n_in,
                              void* d_out, int out_size, void* d_ws, size_t ws_size,
                              hipStream_t stream) {
    const float* feats    = (const float*)d_in[0];
    const int*   src      = (const int*)  d_in[1];
    const int*   dst      = (const int*)  d_in[2];
    const float* W1       = (const float*)d_in[3];
    const float* b1       = (const float*)d_in[4];
    const float* W2       = (const float*)d_in[5];
    const float* b2       = (const float*)d_in[6];
    const float* ln_gamma = (const float*)d_in[7];
    const float* ln_beta  = (const float*)d_in[8];
    const float* dec1_W   = (const float*)d_in[9];
    const float* dec1_b   = (const float*)d_in[10];
    // d_in[11], d_in[12]: dec2 (unused, LAMB_NODE == 0)
    const float* codebook = (const float*)d_in[13];
    float* out = (float*)d_out;

    // ---- workspace carve-out (256B aligned)
    char* base = (char*)d_ws;
    size_t off = 0;
    auto carve = [&](size_t bytes) { char* p = base + off; off = (off + bytes + 255) & ~(size_t)255; return p; };
    float* f_m     = (float*)carve((size_t)MTOT * DD * sizeof(float));  // aggregation buffer
    float* f_h1    = (float*)carve((size_t)MTOT * DD * sizeof(float));  // h1 / layernormed
    float* f_h2    = (float*)carve((size_t)MTOT * DD * sizeof(float));  // h2 == hf
    float* f_q     = (float*)carve((size_t)MTOT * DD * sizeof(float));  // xn, later qe
    float* f_cn    = (float*)carve((size_t)KK   * DD * sizeof(float));  // normalized codebook
    float* f_degout= (float*)carve((size_t)MTOT * sizeof(float));       // -> rs_out
    float* f_rsin  = (float*)carve((size_t)MTOT * sizeof(float));
    int*   i_degin = (int*)  carve((size_t)MTOT * sizeof(int));
    int*   i_rowst = (int*)  carve((size_t)MTOT * sizeof(int));
    int*   i_cursor= (int*)  carve((size_t)MTOT * sizeof(int));
    int*   i_csr   = (int*)  carve((size_t)GG * EE * sizeof(int));
    int*   i_idx   = (int*)  carve((size_t)MTOT * sizeof(int));
    unsigned char* u_adj = (unsigned char*)carve((size_t)GG * NN * NN);
    const long NPARTS = (long)GG * TPG + (MTOT * DD / 256) + 2L * GG + 4;
    float* f_parts = (float*)carve((size_t)NPARTS * sizeof(float));
    float* f_edgepart = f_parts;                        // [GG*TPG]
    float* f_vqpart   = f_parts + (long)GG * TPG;       // [MTOT*DD/256]
    float* f_small    = f_vqpart + (MTOT * DD / 256);
    float* f_ne   = f_small;            // [GG]
    float* f_esum = f_small + GG;       // [GG]
    float* f_sq   = f_small + 2 * GG;   // scalar
    (void)ws_size; (void)in_sizes; (void)n_in; (void)out_size;

    const long ND = (long)MTOT * DD;            // 4,194,304 floats
    const int  GE_BLK = (GG * EE) / 256;        // 1024
    const int  GEMM_BLK = ((MTOT / 32) * 4) / 4;// 512 blocks (128 thr = 4 waves)

    // 1) degrees -> rsqrt(clip(deg,1)); CSR build (deterministic after sort)
    zero_f32_kernel<<<64, 256, 0, stream>>>(f_degout, MTOT);
    zero_f32_kernel<<<64, 256, 0, stream>>>((float*)i_degin, MTOT);
    deg_kernel<<<GE_BLK, 256, 0, stream>>>(src, dst, f_degout, i_degin);
    rs_kernel<<<MTOT / 256, 256, 0, stream>>>(f_degout, MTOT);
    rs_int_kernel<<<MTOT / 256, 256, 0, stream>>>(i_degin, f_rsin, MTOT);
    scan_kernel<<<GG, 256, 0, stream>>>(i_degin, i_rowst, i_cursor);
    csr_fill_kernel<<<GE_BLK, 256, 0, stream>>>(src, dst, i_cursor, i_csr);
    sort_csr_kernel<<<MTOT / 256, 256, 0, stream>>>(i_rowst, i_degin, i_csr);

    // 2) conv1: gather aggregate then WMMA GEMM(+rs_in row scale, bias, relu)
    gather_agg_kernel<<<GG * NN, 256, 0, stream>>>(feats, i_csr, i_rowst, i_degin, f_degout, f_m);
    gemm32x64_kernel<<<GEMM_BLK, 128, 0, stream>>>(f_m, nullptr, f_rsin, W1, b1, f_h1, MTOT, 1);

    // 3) layernorm (in place)
    layernorm_kernel<<<MTOT, 256, 0, stream>>>(f_h1, ln_gamma, ln_beta);

    // 4) conv2
    gather_agg_kernel<<<GG * NN, 256, 0, stream>>>(f_h1, i_csr, i_rowst, i_degin, f_degout, f_m);
    gemm32x64_kernel<<<GEMM_BLK, 128, 0, stream>>>(f_m, nullptr, f_rsin, W2, b2, f_h2, MTOT, 1);

    // 5) cosine VQ: normalize rows, WMMA argmax
    rownorm_kernel<<<MTOT, 256, 0, stream>>>(f_h2, f_q);        // xn
    rownorm_kernel<<<KK, 256, 0, stream>>>(codebook, f_cn);     // cn
    vq_argmax_kernel<<<(MTOT / 32) / 4, 128, 0, stream>>>(f_q, f_cn, i_idx);

    // 6) deterministic partial-sum buffers
    zero_f32_kernel<<<64, 256, 0, stream>>>(f_parts, NPARTS);

    // 7) vq loss partials + reduce
    vq_loss_kernel<<<(int)(ND / 256), 256, 0, stream>>>(f_h2, codebook, i_idx, f_vqpart);
    reduce_sum_kernel<<<1, 256, 0, stream>>>(f_vqpart, ND / 256, f_sq);

    // 8) decoder: qe = codebook[idx] @ dec1_W + dec1_b   (overwrites f_q)
    gemm32x64_kernel<<<GEMM_BLK, 128, 0, stream>>>(codebook, i_idx, nullptr, dec1_W, dec1_b, f_q, MTOT, 0);

    // 9) adjacency + edge counts
    zero_u8_kernel<<<4096, 256, 0, stream>>>(u_adj, (long)GG * NN * NN);
    adj_build_kernel<<<GE_BLK, 256, 0, stream>>>(src, dst, u_adj);
    count_edges_kernel<<<GG * NN, 256, 0, stream>>>(u_adj, f_ne);

    // 10) edge loss (WMMA logits tiles) + per-graph reduce
    edge_loss_kernel<<<(GG * TPG) / 4, 128, 0, stream>>>(f_q, u_adj, f_ne, f_edgepart);
    reduce_graph_kernel<<<GG, 256, 0, stream>>>(f_edgepart, f_esum);

    // 11) combine to scalar
    final_kernel<<<1, 64, 0, stream>>>(f_esum, f_sq, out);
}